// StandardGCN_3994319585553
// MI455X (gfx1250) — compile-verified
//
#include <hip/hip_runtime.h>

typedef __attribute__((ext_vector_type(2))) float v2f;
typedef __attribute__((ext_vector_type(8))) float v8f;
typedef long long i64;

#define FEATS 128
#define LDA 132   // padded LDS stride (132 % 64 == 4) -> conflict-free A reads

// ---------- degree / normalization ----------
__global__ void k_init_deg(float* __restrict__ deg, int n) {
  int i = blockIdx.x * blockDim.x + threadIdx.x;
  if (i < n) deg[i] = 1.0f;  // self-loop weight
}

__global__ void k_deg_accum(const i64* __restrict__ dst, const float* __restrict__ ew,
                            float* __restrict__ deg, int nE) {
  int e = blockIdx.x * blockDim.x + threadIdx.x;
  if (e < nE) atomicAdd(&deg[(size_t)dst[e]], ew[e]);
}

__global__ void k_deg_to_dinv(float* __restrict__ deg, int n) {
  int i = blockIdx.x * blockDim.x + threadIdx.x;
  if (i < n) {
    float d = deg[i];
    deg[i] = (d > 0.0f) ? rsqrtf(d) : 0.0f;
  }
}

// ---------- GEMM: XW = X @ W  (f32 WMMA 16x16x4) ----------
// block = 256 threads = 8 waves; block handles one 16-row M tile;
// wave w computes output columns [16w, 16w+16).
__global__ __launch_bounds__(256) void k_gemm_wmma(
    const float* __restrict__ X, const float* __restrict__ W,
    float* __restrict__ XW, int nRows) {
  __shared__ float ldsA[16 * LDA];
  const int row0 = blockIdx.x * 16;

  // cooperative stage of the 16x128 A tile (coalesced global reads)
  for (int i = threadIdx.x; i < 16 * FEATS; i += 256) {
    int r = i >> 7, c = i & (FEATS - 1);
    int row = row0 + r;
    ldsA[r * LDA + c] = (row < nRows) ? X[(size_t)row * FEATS + c] : 0.0f;
  }
  __syncthreads();

  const int wave = threadIdx.x >> 5;
  const int lane = threadIdx.x & 31;
  const int n0 = wave * 16;
  const int m  = lane & 15;           // A row / B,C column within tile
  const int kh = (lane >> 4) << 1;    // K sub-offset: 0 (lanes 0-15) or 2 (lanes 16-31)

  v8f acc = {};
  for (int k = 0; k < FEATS; k += 4) {
    const int kk = k + kh;
    // A 16x4 f32: VGPR0 = K=kh, VGPR1 = K=kh+1 (per-half-wave)
    v2f a;
    a.x = ldsA[m * LDA + kk];
    a.y = ldsA[m * LDA + kk + 1];
    // B 4x16 f32: row K=kk across lanes (N = n0+m), next VGPR row K=kk+1
    v2f b;
    b.x = W[(size_t)kk * FEATS + n0 + m];
    b.y = W[(size_t)(kk + 1) * FEATS + n0 + m];
    acc = __builtin_amdgcn_wmma_f32_16x16x4_f32(
        /*neg_a=*/false, a, /*neg_b=*/false, b,
        /*c_mod=*/(short)0, acc, /*reuse_a=*/false, /*reuse_b=*/false);
  }

  // C/D 16x16 f32 layout: lane<16 -> M = v, lane>=16 -> M = v+8; N = n0 + (lane&15)
  const int mbase = row0 + ((lane >> 4) << 3);
  const int col = n0 + m;
#pragma unroll
  for (int v = 0; v < 8; ++v) {
    int row = mbase + v;
    if (row < nRows) XW[(size_t)row * FEATS + col] = acc[v];
  }
}

// ---------- self-loop init: agg = xw * dinv^2 ----------
__global__ void k_self_init(const float* __restrict__ xw, const float* __restrict__ dinv,
                            float* __restrict__ agg, int n) {
  int idx = blockIdx.x * blockDim.x + threadIdx.x;  // over n * 32 float4 groups
  if (idx >= n * (FEATS / 4)) return;
  int i = idx >> 5;
  float di = dinv[i];
  float s = di * di;
  float4 v = ((const float4*)xw)[idx];
  float4 o;
  o.x = v.x * s; o.y = v.y * s; o.z = v.z * s; o.w = v.w * s;
  ((float4*)agg)[idx] = o;
}

// ---------- edge scatter-add: one wave32 per edge, float4 per lane ----------
__global__ __launch_bounds__(256) void k_edge_agg(
    const i64* __restrict__ src, const i64* __restrict__ dst,
    const float* __restrict__ ew, const float* __restrict__ dinv,
    const float* __restrict__ xw, float* __restrict__ agg, int nE) {
  int gwave = (int)((blockIdx.x * blockDim.x + threadIdx.x) >> 5);
  int lane = threadIdx.x & 31;
  int nwaves = (int)((gridDim.x * blockDim.x) >> 5);
  for (int e = gwave; e < nE; e += nwaves) {
    size_t s = (size_t)src[e];
    size_t d = (size_t)dst[e];
    float nrm = dinv[s] * ew[e] * dinv[d];
    float4 v = ((const float4*)(xw + s * FEATS))[lane];
    float* ad = agg + d * FEATS + lane * 4;
    atomicAdd(ad + 0, v.x * nrm);
    atomicAdd(ad + 1, v.y * nrm);
    atomicAdd(ad + 2, v.z * nrm);
    atomicAdd(ad + 3, v.w * nrm);
  }
}

// ---------- bias + ReLU (safe in place) ----------
__global__ void k_bias_relu(const float* __restrict__ agg, const float* __restrict__ bias,
                            float* __restrict__ out, int n) {
  int idx = blockIdx.x * blockDim.x + threadIdx.x;
  if (idx >= n * (FEATS / 4)) return;
  int f4 = idx & 31;
  float4 b = ((const float4*)bias)[f4];
  float4 v = ((const float4*)agg)[idx];
  float4 o;
  o.x = fmaxf(v.x + b.x, 0.0f);
  o.y = fmaxf(v.y + b.y, 0.0f);
  o.z = fmaxf(v.z + b.z, 0.0f);
  o.w = fmaxf(v.w + b.w, 0.0f);
  ((float4*)out)[idx] = o;
}

extern "C" void kernel_launch(void* const* d_in, const int* in_sizes, int n_in,
                              void* d_out, int out_size, void* d_ws, size_t ws_size,
                              hipStream_t stream) {
  const float* x  = (const float*)d_in[0];
  const i64*  ei  = (const i64*)d_in[1];   // [2, E] int64
  const float* ew = (const float*)d_in[2];
  const float* W1 = (const float*)d_in[3];
  const float* b1 = (const float*)d_in[4];
  const float* W2 = (const float*)d_in[5];
  const float* b2 = (const float*)d_in[6];
  float* out = (float*)d_out;

  const int N = in_sizes[0] / FEATS;
  const int E = in_sizes[2];
  const i64* src = ei;
  const i64* dst = ei + E;

  char* ws = (char*)d_ws;
  float* dinv = (float*)ws;  // N floats (holds deg, then dinv)
  size_t off = (((size_t)N * 4) + 255) & ~(size_t)255;
  float* bufA = (float*)(ws + off);                                // N*128 f32
  float* bufB = (float*)(ws + off + (size_t)N * FEATS * sizeof(float));

  const dim3 b256(256);
  const int gN    = (N + 255) / 256;
  const int gE    = (E + 255) / 256;
  const int gNF4  = (N * (FEATS / 4) + 255) / 256;
  const int tiles = (N + 15) / 16;
  const int aggBlocks = 4096;  // 32768 waves grid-striding the edge list

  // normalization (shared by both layers)
  k_init_deg<<<gN, b256, 0, stream>>>(dinv, N);
  k_deg_accum<<<gE, b256, 0, stream>>>(dst, ew, dinv, E);
  k_deg_to_dinv<<<gN, b256, 0, stream>>>(dinv, N);

  // layer 1: h1 = relu(norm_agg(x @ W1) + b1)   -> bufA
  k_gemm_wmma<<<tiles, b256, 0, stream>>>(x, W1, bufA, N);
  k_self_init<<<gNF4, b256, 0, stream>>>(bufA, dinv, bufB, N);
  k_edge_agg<<<aggBlocks, b256, 0, stream>>>(src, dst, ew, dinv, bufA, bufB, E);
  k_bias_relu<<<gNF4, b256, 0, stream>>>(bufB, b1, bufA, N);

  // layer 2: out = relu(norm_agg(h1 @ W2) + b2) -> d_out (aggregate in place)
  k_gemm_wmma<<<tiles, b256, 0, stream>>>(bufA, W2, bufB, N);
  k_self_init<<<gNF4, b256, 0, stream>>>(bufB, dinv, out, N);
  k_edge_agg<<<aggBlocks, b256, 0, stream>>>(src, dst, ew, dinv, bufB, out, E);
  k_bias_relu<<<gNF4, b256, 0, stream>>>(out, b2, out, N);
}